// MTANPredictionModel_18829136625683
// MI455X (gfx1250) — compile-verified
//
#include <hip/hip_runtime.h>
#include <hip/hip_bf16.h>
#include <cstdint>

// ---------------------------------------------------------------------------
// MTAN prediction model, fp32 end-to-end, using CDNA5 native fp32 WMMA
// (V_WMMA_F32_16X16X4_F32) for the attention GEMM, the value-projection GEMM
// and the GRU recurrent GEMMs.  Global->LDS staging uses the CDNA5 async
// copy path (GLOBAL_LOAD_ASYNC_TO_LDS_B64 / ASYNCcnt), double-buffered in
// the sequential GRU loop.
//
// Sizes (fixed by the reference):
//   B=8 T=96 N=200 FIN=1 EMB=32 HEADS=1 HID=32 HOR=24 FV=400 HMT=6400
// ---------------------------------------------------------------------------

typedef float v2f __attribute__((ext_vector_type(2)));
typedef float v8f __attribute__((ext_vector_type(8)));

// D = A(16x4 f32) * B(4x16 f32) + C(16x16 f32), full fp32 matrix op.
// 8-arg pattern: (neg_a, A, neg_b, B, c_mod, C, reuse_a, reuse_b)
static __device__ __forceinline__ v8f wmma4(v2f a, v2f b, v8f c) {
  return __builtin_amdgcn_wmma_f32_16x16x4_f32(false, a, false, b,
                                               (short)0, c, false, false);
}

// Async global->LDS copy of 8 bytes (tracked by ASYNCcnt).
static __device__ __forceinline__ void async_copy_b64(float* lds,
                                                      const float* g) {
  uint32_t l = (uint32_t)(uintptr_t)lds;   // low 32 bits = LDS byte offset
  uint64_t ga = (uint64_t)(uintptr_t)g;
  asm volatile("global_load_async_to_lds_b64 %0, %1, off"
               :: "v"(l), "v"(ga)
               : "memory");
}

static __device__ __forceinline__ void wait_async0() {
  asm volatile("s_wait_asynccnt 0x0" ::: "memory");
}

// ---------------------------------------------------------------------------
// Kernel 1: time embedding, q/k projections, scores, row-max, E = exp(s-max)
// One block per batch. LDS: q[96*32], k[96*32], te/s overlapped (<= 60KB).
// ---------------------------------------------------------------------------
__global__ __launch_bounds__(128) void k_scores(
    const float* __restrict__ ts,   // [B,96]
    const float* __restrict__ te_w, // [32,1]
    const float* __restrict__ te_b, // [32]
    const float* __restrict__ q_w,  // [32,32]
    const float* __restrict__ q_b,  // [32]
    const float* __restrict__ k_w,  // [32,32]
    float* __restrict__ E)          // [B,96,96]
{
  __shared__ float sm[15360];          // 60 KB
  float* qs  = sm;                     // [96][32]
  float* ks2 = sm + 3072;              // [96][32]
  float* te  = sm + 6144;              // [96][32]  (overwritten by s later)
  float* s   = sm + 6144;              // [96][96]

  const int b = blockIdx.x;
  const int tid = threadIdx.x;

  for (int i = tid; i < 3072; i += 128) {
    int t = i >> 5, e = i & 31;
    float v = ts[b * 96 + t] * te_w[e] + te_b[e];
    te[t * 32 + e] = (e == 0) ? v : sinf(v);
  }
  __syncthreads();

  for (int i = tid; i < 3072; i += 128) {
    int t = i >> 5, e = i & 31;
    float aq = q_b[e], ak = 0.0f;
    for (int j = 0; j < 32; ++j) {
      float tv = te[t * 32 + j];
      aq += tv * q_w[e * 32 + j];
      ak += tv * k_w[e * 32 + j];
    }
    qs[t * 32 + e] = aq;
    ks2[t * 32 + e] = ak;
  }
  __syncthreads();

  const float scale = 0.17677669529663687f;  // 1/sqrt(32)
  for (int i = tid; i < 9216; i += 128) {
    int tq = i / 96, tk = i % 96;
    float acc = 0.0f;
    for (int j = 0; j < 32; ++j) acc += qs[tq * 32 + j] * ks2[tk * 32 + j];
    s[tq * 96 + tk] = acc * scale;
  }
  __syncthreads();

  if (tid < 96) {
    float mx = -1e30f;
    for (int k = 0; k < 96; ++k) mx = fmaxf(mx, s[tid * 96 + k]);
    for (int k = 0; k < 96; ++k)
      E[(b * 96 + tid) * 96 + k] = __expf(s[tid * 96 + k] - mx);
  }
}

// ---------------------------------------------------------------------------
// Kernel 2: W[b,k,0:400] = m * xm ; W[b,k,400:800] = m   (K-major, coalesced
// as the WMMA B operand).  f = 2n + c : c==0 -> x, c==1 -> mask.
// ---------------------------------------------------------------------------
__global__ void k_buildw(const float* __restrict__ x,
                         const float* __restrict__ mask,
                         float* __restrict__ W)
{
  int idx = blockIdx.x * blockDim.x + threadIdx.x;
  if (idx >= 8 * 96 * 800) return;
  int b = idx / 76800;
  int r = idx % 76800;
  int k = r / 800;
  int col = r % 800;
  int f = (col < 400) ? col : (col - 400);
  int n = f >> 1, c = f & 1;
  float mv = mask[(b * 96 + k) * 200 + n];
  float mb = (mv > 0.5f) ? 1.0f : 0.0f;
  float val;
  if (col < 400) {
    float xmv = (c == 0) ? x[(b * 96 + k) * 200 + n] : mv;
    val = mb * xmv;
  } else {
    val = mb;
  }
  W[idx] = val;
}

// ---------------------------------------------------------------------------
// Kernel 3: per-batch P[96,800] = E[96,96] @ W[96,800] via fp32 WMMA.
// One wave per 16x16 output tile; 24 K-steps.
// ---------------------------------------------------------------------------
__global__ __launch_bounds__(32) void k_attn_gemm(
    const float* __restrict__ E,  // [B,96,96]
    const float* __restrict__ W,  // [B,96,800]
    float* __restrict__ P)        // [B,96,800]
{
  int blk = blockIdx.x;                 // 8 * 6 * 50 = 2400
  int b = blk / 300;
  int rem = blk % 300;
  int mt = rem / 50, nt = rem % 50;
  int lane = threadIdx.x;
  int mn = lane & 15;
  int hi = lane >> 4;
  int kb = hi * 2;

  const float* Eb = E + b * 96 * 96;
  const float* Wb = W + b * 96 * 800;
  int row = mt * 16 + mn;
  int col = nt * 16 + mn;

  v8f acc = {};
  for (int kt = 0; kt < 24; ++kt) {
    int k0 = kt * 4;
    v2f a;
    a.x = Eb[row * 96 + k0 + kb];
    a.y = Eb[row * 96 + k0 + kb + 1];
    v2f bv;
    bv.x = Wb[(k0 + kb) * 800 + col];
    bv.y = Wb[(k0 + kb + 1) * 800 + col];
    acc = wmma4(a, bv, acc);
  }
  float* Pb = P + b * 96 * 800;
#pragma unroll
  for (int r = 0; r < 8; ++r)
    Pb[(mt * 16 + r + hi * 8) * 800 + col] = acc[r];
}

// ---------------------------------------------------------------------------
// Kernel 4: X[row,f] = num/den  (den==0 -> 0; probability ~2^-96)
// ---------------------------------------------------------------------------
__global__ void k_div(const float* __restrict__ P, float* __restrict__ X)
{
  int idx = blockIdx.x * blockDim.x + threadIdx.x;
  if (idx >= 768 * 400) return;
  int row = idx / 400, f = idx % 400;
  float num = P[row * 800 + f];
  float den = P[row * 800 + 400 + f];
  X[idx] = (den > 0.0f) ? (num / den) : 0.0f;
}

// ---------------------------------------------------------------------------
// Kernel 5: transpose v_w [6400,400] -> Vt [400,6400] (coalesced B-loads)
// ---------------------------------------------------------------------------
__global__ void k_transpose(const float* __restrict__ vw, float* __restrict__ Vt)
{
  int idx = blockIdx.x * blockDim.x + threadIdx.x;
  if (idx >= 400 * 6400) return;
  int k = idx / 6400, nn = idx % 6400;
  Vt[idx] = vw[nn * 400 + k];
}

// ---------------------------------------------------------------------------
// Kernel 6: Y[768,6400] = X[768,400] @ Vt[400,6400] + v_b.
// 128-thread blocks (4 waves).  The 16x400 A tile is staged once into LDS
// via async global->LDS copies and shared by all 4 waves (each computing a
// 16x64 strip).  LDS row stride 404 (mod 64 = 20) is bank-conflict free.
// ---------------------------------------------------------------------------
__global__ __launch_bounds__(128) void k_vproj(
    const float* __restrict__ X,   // [768,400]
    const float* __restrict__ Vt,  // [400,6400]
    const float* __restrict__ vb,  // [6400]
    float* __restrict__ Y)         // [768,6400]
{
  __shared__ float As[16][404];

  const int tid = threadIdx.x;
  const int wave = tid >> 5;
  const int lane = tid & 31;
  const int mt = blockIdx.x / 25;   // 48 M tiles
  const int ng = blockIdx.x % 25;   // 25 N groups of 256

  // async-stage A tile: rows mt*16 .. mt*16+15, 400 floats each
  for (int i = tid; i < 3200; i += 128) {
    int r = i / 200;
    int c2 = (i % 200) * 2;
    const float* g = X + (size_t)(mt * 16 + r) * 400 + c2;
    async_copy_b64(&As[r][c2], g);
  }
  wait_async0();
  __syncthreads();

  const int mn = lane & 15;
  const int hi = lane >> 4;
  const int kb = hi * 2;
  const int n0 = ng * 256 + wave * 64;

  v8f acc[4];
#pragma unroll
  for (int j = 0; j < 4; ++j) acc[j] = (v8f){};

  for (int kt = 0; kt < 100; ++kt) {
    int k0 = kt * 4;
    v2f a;
    a.x = As[mn][k0 + kb];
    a.y = As[mn][k0 + kb + 1];
    const float* B0 = Vt + (size_t)(k0 + kb) * 6400;
    const float* B1 = Vt + (size_t)(k0 + kb + 1) * 6400;
    if (kt < 96) {  // prefetch the B rows 4 k-steps ahead
      __builtin_prefetch(B0 + 16 * 6400 + n0, 0, 0);
    }
#pragma unroll
    for (int j = 0; j < 4; ++j) {
      v2f bv;
      bv.x = B0[n0 + j * 16 + mn];
      bv.y = B1[n0 + j * 16 + mn];
      acc[j] = wmma4(a, bv, acc[j]);
    }
  }
#pragma unroll
  for (int j = 0; j < 4; ++j) {
    int col = n0 + j * 16 + mn;
#pragma unroll
    for (int r = 0; r < 8; ++r) {
      int m = mt * 16 + r + hi * 8;
      Y[(size_t)m * 6400 + col] = acc[j][r] + vb[col];
    }
  }
}

// ---------------------------------------------------------------------------
// Kernel 7: GRU over T=96 steps. 100 blocks x 16 rows; rows are independent.
// Weights (transposed) + h + gate buffers live in LDS.  The per-step x tile
// is DOUBLE-BUFFERED: step t+1's tile is fetched with async global->LDS
// copies while step t's WMMAs and gate update run (ASYNCcnt hides latency).
// ---------------------------------------------------------------------------
__global__ __launch_bounds__(128) void k_gru(
    const float* __restrict__ Y,     // [768,6400] -> xs[t][row][32]
    const float* __restrict__ wih,   // [96,32]
    const float* __restrict__ whh,   // [96,32]
    const float* __restrict__ bih,   // [96]
    const float* __restrict__ bhh,   // [96]
    float* __restrict__ hout)        // [1600,32]
{
  __shared__ float wihT[32][96];
  __shared__ float whhT[32][96];
  __shared__ float hs[16][32];
  __shared__ float xs[2][16][32];
  __shared__ float GI[16][96];
  __shared__ float GH[16][96];

  const int tid = threadIdx.x;
  const int wave = tid >> 5;
  const int lane = tid & 31;
  const int mn = lane & 15;
  const int hi = lane >> 4;
  const int kb = hi * 2;
  const int rowBase = blockIdx.x * 16;

  for (int i = tid; i < 96 * 32; i += 128) {
    int ncol = i / 32, k = i % 32;
    wihT[k][ncol] = wih[ncol * 32 + k];
    whhT[k][ncol] = whh[ncol * 32 + k];
  }
  for (int i = tid; i < 512; i += 128) hs[i >> 5][i & 31] = 0.0f;

  // issue async copy of the t=0 x tile (256 x b64 chunks)
  auto issue_x = [&](int t, int buf) {
    for (int i = tid; i < 256; i += 128) {
      int r = i >> 4;             // floats 2i..2i+1 -> row = 2i/32
      int j = (i & 15) * 2;
      int grow = rowBase + r;
      int b = grow / 200, n = grow % 200;
      const float* g = Y + (size_t)(b * 96 + t) * 6400 + n * 32 + j;
      async_copy_b64(&xs[buf][r][j], g);
    }
  };
  issue_x(0, 0);

  for (int t = 0; t < 96; ++t) {
    const int cur = t & 1;
    wait_async0();        // my async chunks for xs[cur] are in LDS
    __syncthreads();      // everyone's are

    if (t + 1 < 96) issue_x(t + 1, cur ^ 1);   // prefetch next step's tile

    // 12 WMMA tile-jobs: jobs 0..5 -> GI n-tiles, 6..11 -> GH n-tiles
    for (int job = wave; job < 12; job += 4) {
      int nt = job % 6;
      bool isGh = (job >= 6);
      const float(*Wt)[96] = isGh ? whhT : wihT;
      const float(*A)[32] = isGh ? hs : xs[cur];
      v8f acc = {};
#pragma unroll
      for (int kt = 0; kt < 8; ++kt) {
        int k0 = kt * 4;
        v2f a;
        a.x = A[mn][k0 + kb];
        a.y = A[mn][k0 + kb + 1];
        v2f bv;
        bv.x = Wt[k0 + kb][nt * 16 + mn];
        bv.y = Wt[k0 + kb + 1][nt * 16 + mn];
        acc = wmma4(a, bv, acc);
      }
      float(*G)[96] = isGh ? GH : GI;
#pragma unroll
      for (int r = 0; r < 8; ++r) G[r + hi * 8][nt * 16 + mn] = acc[r];
    }
    __syncthreads();

    // gate update: h' = (1-z)*n + z*h
    for (int i = tid; i < 512; i += 128) {
      int r = i >> 5, j = i & 31;
      float ir = GI[r][j] + bih[j];
      float hr = GH[r][j] + bhh[j];
      float iz = GI[r][32 + j] + bih[32 + j];
      float hz = GH[r][32 + j] + bhh[32 + j];
      float in_ = GI[r][64 + j] + bih[64 + j];
      float hn = GH[r][64 + j] + bhh[64 + j];
      float rg = 1.0f / (1.0f + __expf(-(ir + hr)));
      float zg = 1.0f / (1.0f + __expf(-(iz + hz)));
      float ng = tanhf(in_ + rg * hn);
      hs[r][j] = (1.0f - zg) * ng + zg * hs[r][j];
    }
    __syncthreads();
  }

  for (int i = tid; i < 512; i += 128) {
    int r = i >> 5, j = i & 31;
    hout[(size_t)(rowBase + r) * 32 + j] = hs[r][j];
  }
}

// ---------------------------------------------------------------------------
// Kernel 8: decoder.  hh = relu(h@mlp_w^T + b); y = hh@out_w^T + b;
// out[b,hor,n,0].  One 32-lane wave per row.
// ---------------------------------------------------------------------------
__global__ __launch_bounds__(32) void k_dec(
    const float* __restrict__ hfin,  // [1600,32]
    const float* __restrict__ mlp_w, // [32,32]
    const float* __restrict__ mlp_b, // [32]
    const float* __restrict__ out_w, // [24,32]
    const float* __restrict__ out_b, // [24]
    float* __restrict__ out)         // [8,24,200]
{
  __shared__ float hh[32];
  int row = blockIdx.x;
  int j = threadIdx.x;
  const float* h = hfin + (size_t)row * 32;
  float acc = mlp_b[j];
  for (int k = 0; k < 32; ++k) acc += h[k] * mlp_w[j * 32 + k];
  hh[j] = fmaxf(acc, 0.0f);
  __syncthreads();
  if (j < 24) {
    float y = out_b[j];
    for (int k = 0; k < 32; ++k) y += hh[k] * out_w[j * 32 + k];
    int b = row / 200, n = row % 200;
    out[(size_t)(b * 24 + j) * 200 + n] = y;
  }
}

// ---------------------------------------------------------------------------
// Workspace layout (floats).  Total = 9,136,128 floats = 36.5 MB.
// ---------------------------------------------------------------------------
static const size_t OFF_E = 0;             // 8*96*96      = 73728
static const size_t OFF_W = 73728;         // 8*96*800     = 614400
static const size_t OFF_P = 688128;        // 8*96*800     = 614400
static const size_t OFF_X = 1302528;       // 768*400      = 307200
static const size_t OFF_VT = 1609728;      // 400*6400     = 2560000
static const size_t OFF_Y = 4169728;       // 768*6400     = 4915200
static const size_t OFF_H = 9084928;       // 1600*32      = 51200

extern "C" void kernel_launch(void* const* d_in, const int* in_sizes, int n_in,
                              void* d_out, int out_size, void* d_ws, size_t ws_size,
                              hipStream_t stream) {
  const float* x        = (const float*)d_in[0];
  // d_in[1] = u (unused by the model)
  const float* mask     = (const float*)d_in[2];
  const float* tsteps   = (const float*)d_in[3];
  const float* te_w     = (const float*)d_in[4];
  const float* te_b     = (const float*)d_in[5];
  const float* q_w      = (const float*)d_in[6];
  const float* q_b      = (const float*)d_in[7];
  const float* k_w      = (const float*)d_in[8];
  const float* v_w      = (const float*)d_in[9];
  const float* v_b      = (const float*)d_in[10];
  const float* gru_wih  = (const float*)d_in[11];
  const float* gru_whh  = (const float*)d_in[12];
  const float* gru_bih  = (const float*)d_in[13];
  const float* gru_bhh  = (const float*)d_in[14];
  const float* mlp_w    = (const float*)d_in[15];
  const float* mlp_b    = (const float*)d_in[16];
  const float* out_w    = (const float*)d_in[17];
  const float* out_b    = (const float*)d_in[18];

  float* ws = (float*)d_ws;
  float* E   = ws + OFF_E;
  float* W   = ws + OFF_W;
  float* P   = ws + OFF_P;
  float* X   = ws + OFF_X;
  float* Vt  = ws + OFF_VT;
  float* Y   = ws + OFF_Y;
  float* Hf  = ws + OFF_H;
  float* out = (float*)d_out;

  k_scores<<<8, 128, 0, stream>>>(tsteps, te_w, te_b, q_w, q_b, k_w, E);
  k_buildw<<<(8 * 96 * 800 + 255) / 256, 256, 0, stream>>>(x, mask, W);
  k_attn_gemm<<<2400, 32, 0, stream>>>(E, W, P);
  k_div<<<(768 * 400 + 255) / 256, 256, 0, stream>>>(P, X);
  k_transpose<<<(400 * 6400 + 255) / 256, 256, 0, stream>>>(v_w, Vt);
  k_vproj<<<1200, 128, 0, stream>>>(X, Vt, v_b, Y);
  k_gru<<<100, 128, 0, stream>>>(Y, gru_wih, gru_whh, gru_bih, gru_bhh, Hf);
  k_dec<<<1600, 32, 0, stream>>>(Hf, mlp_w, mlp_b, out_w, out_b, out);
}